// TemporalTransELite_41781441855720
// MI455X (gfx1250) — compile-verified
//
#include <hip/hip_runtime.h>
#include <stdint.h>

// TemporalTransE scoring: out[b,n] = -sum_d |q[b,d] - E[n,d]|
//   q[b] = E[h[b]] + R[r[b]] + T[t[b]]
// B = 128 queries, D = 128 dims, N = 200000 entities.
//
// MI455X / gfx1250 (wave32) strategy:
//  - L1 distance is not bilinear -> WMMA does not apply; VALU f32 kernel
//    (v_sub_f32 + v_add_f32 with |src| modifier, VOPD dual-issuable).
//  - Entity tile staged into LDS *transposed* with global_load_async_to_lds_b32
//    (per-lane LDS addresses give transpose-on-copy), ASYNCcnt-tracked.
//  - Query table stored d-major; q rows read as uniform float4 -> s_load_b128+
//    (scalar pipe, co-issues with VALU).

#define DIMS 128
#define NQ   128
#define NT   32   // entities per block (one wave32)

typedef float v4f __attribute__((ext_vector_type(4)));

__global__ __launch_bounds__(32) void score_kernel(
    const float* __restrict__ E,
    const float* __restrict__ qT,   // [DIMS][NQ] d-major
    float* __restrict__ out,        // [NQ][N]
    int N) {
  __shared__ float eT[DIMS * NT];   // eT[d*NT + n] = E[n0+n][d]  (16 KB)

  const unsigned lane = threadIdx.x;         // 0..31
  const long     n0   = (long)blockIdx.x * NT;

  // ---- Stage entity tile into LDS, transposed, via async copies ----------
  {
    const unsigned ldsbase = (unsigned)(uintptr_t)(void*)eT;
    const unsigned long long sbase =
        (unsigned long long)(uintptr_t)(E + n0 * DIMS);
    #pragma unroll
    for (int dblk = 0; dblk < 4; ++dblk) {
      const unsigned d = (unsigned)(dblk * 32) + lane;
      #pragma unroll 4
      for (int n = 0; n < NT; ++n) {
        const unsigned lds_addr = ldsbase + (d * NT + (unsigned)n) * 4u;
        const unsigned voff     = ((unsigned)n * DIMS + d) * 4u;
        asm volatile("global_load_async_to_lds_b32 %0, %1, %2"
                     :
                     : "v"(lds_addr), "v"(voff), "s"(sbase)
                     : "memory");
      }
    }
    asm volatile("s_wait_asynccnt 0" ::: "memory");
    // One wave per workgroup: wave reads only LDS it wrote; no barrier needed.
  }

  // ---- Score: lane owns entity (n0+lane); 4 chunks of 32 queries ---------
  for (int c = 0; c < 4; ++c) {
    float acc[32];
    #pragma unroll
    for (int i = 0; i < 32; ++i) acc[i] = 0.0f;

    const float* __restrict__ qc = qT + c * 32;  // uniform base for chunk
    #pragma unroll 2
    for (int d = 0; d < DIMS; ++d) {
      const float ed = eT[d * NT + lane];        // conflict-free ds_load_b32
      // 32 uniform q values as 8 x float4 -> wide s_load on the scalar pipe.
      const v4f* __restrict__ qrow = (const v4f*)(qc + d * NQ);
      #pragma unroll
      for (int v = 0; v < 8; ++v) {
        const v4f q4 = qrow[v];
        acc[v * 4 + 0] += __builtin_fabsf(q4.x - ed);
        acc[v * 4 + 1] += __builtin_fabsf(q4.y - ed);
        acc[v * 4 + 2] += __builtin_fabsf(q4.z - ed);
        acc[v * 4 + 3] += __builtin_fabsf(q4.w - ed);
      }
    }

    #pragma unroll
    for (int bb = 0; bb < 32; ++bb) {
      out[(long)(c * 32 + bb) * N + n0 + lane] = -acc[bb];
    }
  }
}

__global__ __launch_bounds__(128) void build_query_kernel(
    const float* __restrict__ E, const float* __restrict__ R,
    const float* __restrict__ T,
    const int* __restrict__ h, const int* __restrict__ r,
    const int* __restrict__ t,
    float* __restrict__ qT /* [DIMS][NQ], d-major */) {
  const int b = blockIdx.x;    // 0..127
  const int d = threadIdx.x;   // 0..127
  const float v = E[(long)h[b] * DIMS + d]
                + R[(long)r[b] * DIMS + d]
                + T[(long)t[b] * DIMS + d];
  qT[(long)d * NQ + b] = v;
}

extern "C" void kernel_launch(void* const* d_in, const int* in_sizes, int n_in,
                              void* d_out, int out_size, void* d_ws,
                              size_t ws_size, hipStream_t stream) {
  const float* E = (const float*)d_in[0];   // [N, 128]
  const float* R = (const float*)d_in[1];   // [500, 128]
  const float* T = (const float*)d_in[2];   // [365, 128]
  const int*   h = (const int*)d_in[3];     // [128]
  const int*   r = (const int*)d_in[4];     // [128]
  const int*   t = (const int*)d_in[5];     // [128]
  float* out = (float*)d_out;               // [128, N]
  float* qT  = (float*)d_ws;                // [128][128] d-major (64 KB)

  const int N = in_sizes[0] / DIMS;         // 200000 (divisible by NT=32)

  build_query_kernel<<<NQ, DIMS, 0, stream>>>(E, R, T, h, r, t, qT);
  score_kernel<<<N / NT, NT, 0, stream>>>(E, qT, out, N);
}